// KGTN_60524679135271
// MI455X (gfx1250) — compile-verified
//
#include <hip/hip_runtime.h>
#include <hip/hip_bf16.h>

// ---------------------------------------------------------------------------
// KGTN: 3-step GGNN over classifier weights + final x @ W^T classifier.
// All GEMMs run on V_WMMA_F32_16X16X4_F32 (fp32 matrix path, wave32).
// BK=32 slabs amortize barriers; global_prefetch hides next-slab latency.
// ---------------------------------------------------------------------------

typedef __attribute__((ext_vector_type(2))) float v2f;
typedef __attribute__((ext_vector_type(8))) float v8f;

#define GEMM_BM 128
#define GEMM_BN 64
#define GEMM_BK 32
#define GEMM_BKP 34   // padded LDS K-stride (even -> keeps b64 alignment of (k,k+1) pairs)

// C[M,Nc] = op(A)[M,K] @ B[K,Nc] (+ C if accum) (+ bias[col] if bias != null)
// op(A) = A            when transA == 0  (A stored row-major, leading dim lda)
// op(A) = A^T          when transA == 1  (A stored [K, M] row-major, lda = row length)
// M multiple of 128, Nc multiple of 64, K multiple of 32.
__global__ __launch_bounds__(256)
void wmma_gemm_f32(const float* __restrict__ A, const float* __restrict__ B,
                   const float* __restrict__ bias, float* __restrict__ C,
                   int M, int Nc, int K, int lda, int ldb, int ldc,
                   int transA, int accum)
{
    __shared__ float As[GEMM_BM * GEMM_BKP];      // As[r*BKP + k] = op(A)[blockM+r, kt+k]
    __shared__ float Bs[GEMM_BN * GEMM_BKP];      // Bs[n*BKP + k] = B[kt+k, blockN+n]

    const int tid  = threadIdx.x;
    const int lane = tid & 31;
    const int wave = tid >> 5;        // 0..7
    const int lo   = lane & 15;
    const int hi   = lane >> 4;       // 0 or 1

    const int wm = wave & 3;          // wave row  -> 32-row slab
    const int wn = wave >> 2;         // wave col  -> 32-col slab

    const int blockM = blockIdx.x * GEMM_BM;
    const int blockN = blockIdx.y * GEMM_BN;

    v8f acc[2][2] = {};               // 32x32 macro tile per wave

    for (int kt = 0; kt < K; kt += GEMM_BK) {
        // ---- prefetch next K-slab while this one is being staged/consumed ----
        const int ktn = kt + GEMM_BK;
        if (ktn < K) {
            if (!transA) {
                for (int i = tid; i < (GEMM_BM * GEMM_BK) / 16; i += 256) {
                    int r   = i >> 1;                 // 2 cache chunks (16 floats) per row
                    int k16 = (i & 1) * 16;
                    __builtin_prefetch(A + (size_t)(blockM + r) * lda + ktn + k16, 0, 3);
                }
            } else {
                for (int i = tid; i < (GEMM_BK * GEMM_BM) / 16; i += 256) {
                    int k   = i >> 3;                 // 8 chunks of 16 floats per k-row
                    int r16 = (i & 7) * 16;
                    __builtin_prefetch(A + (size_t)(ktn + k) * lda + blockM + r16, 0, 3);
                }
            }
            for (int i = tid; i < (GEMM_BK * GEMM_BN) / 16; i += 256) {
                int k   = i >> 2;                     // 4 chunks of 16 floats per k-row
                int n16 = (i & 3) * 16;
                __builtin_prefetch(B + (size_t)(ktn + k) * ldb + blockN + n16, 0, 3);
            }
        }

        // ---- stage op(A) tile: 128 x 32 ----
        if (!transA) {
            for (int i = tid; i < (GEMM_BM * GEMM_BK) / 4; i += 256) {
                int r  = i >> 3;                      // 8 float4 per row
                int k4 = (i & 7) * 4;
                const float4 v = *(const float4*)(A + (size_t)(blockM + r) * lda + kt + k4);
                float* dst = &As[r * GEMM_BKP + k4];
                dst[0] = v.x; dst[1] = v.y; dst[2] = v.z; dst[3] = v.w;
            }
        } else {
            // A stored [K, M]: k-rows contiguous along r
            for (int i = tid; i < (GEMM_BK * GEMM_BM) / 4; i += 256) {
                int k  = i >> 5;                      // 32 float4 per k-row
                int r4 = (i & 31) * 4;
                const float4 v = *(const float4*)(A + (size_t)(kt + k) * lda + blockM + r4);
                As[(r4 + 0) * GEMM_BKP + k] = v.x;
                As[(r4 + 1) * GEMM_BKP + k] = v.y;
                As[(r4 + 2) * GEMM_BKP + k] = v.z;
                As[(r4 + 3) * GEMM_BKP + k] = v.w;
            }
        }
        // ---- stage B tile transposed: Bs[n][k] = B[kt+k][blockN+n] ----
        for (int i = tid; i < (GEMM_BK * GEMM_BN) / 4; i += 256) {
            int k  = i >> 4;                          // 16 float4 per k-row
            int n4 = (i & 15) * 4;
            const float4 v = *(const float4*)(B + (size_t)(kt + k) * ldb + blockN + n4);
            Bs[(n4 + 0) * GEMM_BKP + k] = v.x;
            Bs[(n4 + 1) * GEMM_BKP + k] = v.y;
            Bs[(n4 + 2) * GEMM_BKP + k] = v.z;
            Bs[(n4 + 3) * GEMM_BKP + k] = v.w;
        }
        __syncthreads();

        // ---- 16x16x4 WMMA over the BK=32 slab (32 wmma / wave / slab) ----
        #pragma unroll
        for (int k0 = 0; k0 < GEMM_BK; k0 += 4) {
            const int kk = k0 + hi * 2;   // lanes 0-15: K=k0,k0+1; lanes 16-31: K=k0+2,k0+3
            #pragma unroll
            for (int sm = 0; sm < 2; ++sm) {
                const int row = wm * 32 + sm * 16 + lo;
                const v2f a = *(const v2f*)&As[row * GEMM_BKP + kk];
                #pragma unroll
                for (int sn = 0; sn < 2; ++sn) {
                    const int col = wn * 32 + sn * 16 + lo;
                    const v2f b = *(const v2f*)&Bs[col * GEMM_BKP + kk];
                    acc[sm][sn] = __builtin_amdgcn_wmma_f32_16x16x4_f32(
                        false, a, false, b, (short)0, acc[sm][sn], false, false);
                }
            }
        }
        __syncthreads();
    }

    // ---- epilogue: VGPR i of C frag holds (M = i + 8*hi, N = lo) ----
    #pragma unroll
    for (int sm = 0; sm < 2; ++sm) {
        #pragma unroll
        for (int sn = 0; sn < 2; ++sn) {
            const int col = blockN + wn * 32 + sn * 16 + lo;
            const float bv = bias ? bias[col] : 0.0f;
            #pragma unroll
            for (int i = 0; i < 8; ++i) {
                const int row = blockM + wm * 32 + sm * 16 + hi * 8 + i;
                float* p = C + (size_t)row * ldc + col;
                float v = acc[sm][sn][i] + bv;
                if (accum) v += *p;
                *p = v;
            }
        }
    }
}

// out[cols, rows] = in[rows, cols]^T  (32x32 LDS tiles, block 32x8)
__global__ __launch_bounds__(256)
void transpose_f32(const float* __restrict__ in, float* __restrict__ out,
                   int rows, int cols)
{
    __shared__ float tile[32][33];
    const int bx = blockIdx.x * 32;   // over cols
    const int by = blockIdx.y * 32;   // over rows
    const int tx = threadIdx.x, ty = threadIdx.y;
    #pragma unroll
    for (int j = 0; j < 32; j += 8)
        tile[ty + j][tx] = in[(size_t)(by + ty + j) * cols + bx + tx];
    __syncthreads();
    #pragma unroll
    for (int j = 0; j < 32; j += 8)
        out[(size_t)(bx + ty + j) * rows + by + tx] = tile[tx][ty + j];
}

__device__ __forceinline__ float sigmoid_f(float x) {
    return 1.0f / (1.0f + __expf(-x));
}

// rn = sigmoid(Rp) * nodes
__global__ void rn_kernel(const float* __restrict__ Rp, const float* __restrict__ nodes,
                          float* __restrict__ rn, int n)
{
    for (int i = blockIdx.x * blockDim.x + threadIdx.x; i < n; i += gridDim.x * blockDim.x)
        rn[i] = sigmoid_f(Rp[i]) * nodes[i];
}

// nodes = (1 - sigmoid(Zp)) * nodes + sigmoid(Zp) * tanh(Hp)
__global__ void gru_update_kernel(const float* __restrict__ Zp, const float* __restrict__ Hp,
                                  float* __restrict__ nodes, int n)
{
    for (int i = blockIdx.x * blockDim.x + threadIdx.x; i < n; i += gridDim.x * blockDim.x) {
        const float z = sigmoid_f(Zp[i]);
        const float h = tanhf(Hp[i]);
        nodes[i] = (1.0f - z) * nodes[i] + z * h;
    }
}

// deterministic two-stage sum of squares
__global__ __launch_bounds__(256)
void l2_partial_kernel(const float* __restrict__ s, float* __restrict__ partials, int n)
{
    __shared__ float red[256];
    float acc = 0.0f;
    for (int i = blockIdx.x * blockDim.x + threadIdx.x; i < n; i += gridDim.x * blockDim.x) {
        const float v = s[i];
        acc += v * v;
    }
    red[threadIdx.x] = acc;
    __syncthreads();
    for (int off = 128; off > 0; off >>= 1) {
        if (threadIdx.x < off) red[threadIdx.x] += red[threadIdx.x + off];
        __syncthreads();
    }
    if (threadIdx.x == 0) partials[blockIdx.x] = red[0];
}

__global__ __launch_bounds__(256)
void l2_final_kernel(const float* __restrict__ partials, float* __restrict__ out)
{
    __shared__ float red[256];
    red[threadIdx.x] = partials[threadIdx.x];
    __syncthreads();
    for (int off = 128; off > 0; off >>= 1) {
        if (threadIdx.x < off) red[threadIdx.x] += red[threadIdx.x + off];
        __syncthreads();
    }
    if (threadIdx.x == 0) *out = red[0];
}

// ---------------------------------------------------------------------------

static inline void launch_gemm(const float* A, const float* B, const float* bias,
                               float* C, int M, int Nc, int K,
                               int lda, int ldb, int ldc, int transA, int accum,
                               hipStream_t stream)
{
    dim3 grid(M / GEMM_BM, Nc / GEMM_BN);
    wmma_gemm_f32<<<grid, dim3(256), 0, stream>>>(A, B, bias, C, M, Nc, K,
                                                  lda, ldb, ldc, transA, accum);
}

extern "C" void kernel_launch(void* const* d_in, const int* in_sizes, int n_in,
                              void* d_out, int out_size, void* d_ws, size_t ws_size,
                              hipStream_t stream)
{
    (void)in_sizes; (void)n_in; (void)out_size; (void)ws_size;

    constexpr int H  = 1024;           // FEATURE_DIM
    constexpr int N  = 2048;           // NUM_CLASSES
    constexpr int B  = 8192;           // BATCH
    constexpr int NH = N * H;          // node matrix elements

    const float* x    = (const float*)d_in[0];    // [B, H]
    const float* lfw  = (const float*)d_in[1];    // [H, N]
    const float* inM  = (const float*)d_in[2];    // [N, N]
    const float* w3w  = (const float*)d_in[3];    // [2H, H]
    const float* w3u  = (const float*)d_in[4];    // [H, H]
    const float* w4w  = (const float*)d_in[5];    // [2H, H]
    const float* w4u  = (const float*)d_in[6];    // [H, H]
    const float* w5w  = (const float*)d_in[7];    // [2H, H]
    const float* w5u  = (const float*)d_in[8];    // [H, H]
    const float* fcw  = (const float*)d_in[9];    // [2H, H]
    const float* fcb  = (const float*)d_in[10];   // [H]

    float* out = (float*)d_out;                   // [B*N] output, then l2 scalar

    // workspace layout: 7 x [N,H] fp32 buffers + 256 reduction partials (~59 MB)
    float* ws       = (float*)d_ws;
    float* nodes    = ws + (size_t)0 * NH;
    float* initN    = ws + (size_t)1 * NH;
    float* A1       = ws + (size_t)2 * NH;   // in_matrix @ nodes; later reused as rn
    float* A2       = ws + (size_t)3 * NH;   // in_matrix^T @ nodes
    float* Zp       = ws + (size_t)4 * NH;   // z pre-act; later reused as S (step_out)
    float* Rp       = ws + (size_t)5 * NH;   // r pre-act; later reused as W^T
    float* Hp       = ws + (size_t)6 * NH;   // h pre-act
    float* partials = ws + (size_t)7 * NH;   // 256 floats

    const dim3 eb(256), eg(2048);            // grid-stride elementwise config

    // init_nodes = nodes = last_fc_weight^T  ([H,N] -> [N,H])
    {
        dim3 g(N / 32, H / 32), b(32, 8);
        transpose_f32<<<g, b, 0, stream>>>(lfw, initN, H, N);
        transpose_f32<<<g, b, 0, stream>>>(lfw, nodes, H, N);
    }

    for (int t = 0; t < 3; ++t) {
        // aggregation
        launch_gemm(inM, nodes, nullptr, A1, N, H, N, N, H, H, 0, 0, stream);  // in @ nodes
        launch_gemm(inM, nodes, nullptr, A2, N, H, N, N, H, H, 1, 0, stream);  // in^T @ nodes

        // z gate pre-activation:  Zp = A1@w3w_top + A2@w3w_bot + nodes@w3u
        launch_gemm(A1,    w3w,                 nullptr, Zp, N, H, H, H, H, H, 0, 0, stream);
        launch_gemm(A2,    w3w + (size_t)H * H, nullptr, Zp, N, H, H, H, H, H, 0, 1, stream);
        launch_gemm(nodes, w3u,                 nullptr, Zp, N, H, H, H, H, H, 0, 1, stream);

        // r gate pre-activation
        launch_gemm(A1,    w4w,                 nullptr, Rp, N, H, H, H, H, H, 0, 0, stream);
        launch_gemm(A2,    w4w + (size_t)H * H, nullptr, Rp, N, H, H, H, H, H, 0, 1, stream);
        launch_gemm(nodes, w4u,                 nullptr, Rp, N, H, H, H, H, H, 0, 1, stream);

        // candidate pre-activation (A1/A2 parts first, then rn part)
        launch_gemm(A1, w5w,                  nullptr, Hp, N, H, H, H, H, H, 0, 0, stream);
        launch_gemm(A2, w5w + (size_t)H * H,  nullptr, Hp, N, H, H, H, H, H, 0, 1, stream);
        rn_kernel<<<eg, eb, 0, stream>>>(Rp, nodes, A1, NH);   // A1 := sigmoid(Rp)*nodes
        launch_gemm(A1, w5u,                  nullptr, Hp, N, H, H, H, H, H, 0, 1, stream);

        // GRU update
        gru_update_kernel<<<eg, eb, 0, stream>>>(Zp, Hp, nodes, NH);
    }

    // step_out S = nodes @ fcw_top + init_nodes @ fcw_bot + b   (S reuses Zp)
    launch_gemm(nodes, fcw,                 fcb,     Zp, N, H, H, H, H, H, 0, 0, stream);
    launch_gemm(initN, fcw + (size_t)H * H, nullptr, Zp, N, H, H, H, H, H, 0, 1, stream);

    // weight = S^T  ([N,H] -> [H,N], reuses Rp)
    {
        dim3 g(H / 32, N / 32), b(32, 8);
        transpose_f32<<<g, b, 0, stream>>>(Zp, Rp, N, H);
    }

    // output = x @ weight   [B,H] @ [H,N] -> [B,N]
    launch_gemm(x, Rp, nullptr, out, B, N, H, H, N, N, 0, 0, stream);

    // l2_reg = sum(weight^2) = sum(S^2)  (deterministic two-stage reduction)
    l2_partial_kernel<<<256, 256, 0, stream>>>(Zp, partials, NH);
    l2_final_kernel<<<1, 256, 0, stream>>>(partials, out + (size_t)B * N);
}